// StructNConv2D_gx_with_d_22660247454231
// MI455X (gfx1250) — compile-verified
//
#include <hip/hip_runtime.h>
#include <hip/hip_bf16.h>

// ---------------------------------------------------------------------------
// Fully fused StructNConv2D_gx_with_d for MI455X (gfx1250, wave32).
//   B=8, C=O=32, H=W=256, K=5, stride=1, pad=2, dil=1
// One kernel, one HBM pass: ~268MB in + 134MB out => ~22.5us roofline floor.
// 1x1 channel conv (per-pixel 32x32 GEMM, ~2.1 GFLOP) runs on the matrix
// units via v_wmma_f32_16x16x4_f32 (fp32-exact, matches reference dtype).
// ---------------------------------------------------------------------------

#define B_    8
#define C_IN  32
#define C_OUT 32
#define H_    256
#define W_    256
#define TH    8           // tile height
#define TW    16          // tile width
#define HALO_H (TH + 4)   // 12
#define HALO_W (TW + 4)   // 20
#define HALO_N (HALO_H * HALO_W)  // 240
#define XSTRIDE 258       // 256 cols used; stride!=0 mod 64 banks
#define EPSF 1e-20f

typedef float v2f __attribute__((ext_vector_type(2)));
typedef float v8f __attribute__((ext_vector_type(8)));

__device__ __forceinline__ float softplus_f(float x) {
    // log1p(exp(x)), numerically stable
    return fmaxf(x, 0.0f) + log1pf(expf(-fabsf(x)));
}

__launch_bounds__(256)
__global__ void structnconv_gx_fused(const float* __restrict__ d,
                                     const float* __restrict__ cd,
                                     const float* __restrict__ gx,
                                     const float* __restrict__ cgx,
                                     const float* __restrict__ w_prop,
                                     const float* __restrict__ spatial_weight,
                                     const float* __restrict__ channel_weight,
                                     const float* __restrict__ bias,
                                     float* __restrict__ out) {
    // ---- dynamic LDS layout -------------------------------------------------
    extern __shared__ float smem[];
    float* haloA = smem;                           // [32][240] cgx_prop*gx_prop
    float* haloC = haloA + C_IN * HALO_N;          // [32][240] cgx_prop
    float* X     = haloC + C_IN * HALO_N;          // [32][XSTRIDE] cols 0..127 = cgx_s*gx_s, 128..255 = cgx_s

    __shared__ float sw_s[C_IN * 25];              // softplus(spatial_weight)
    __shared__ float cw_s[C_OUT * 33];             // softplus(channel_weight), row stride 33
    __shared__ float wp_s[C_IN];
    __shared__ float bias_s[C_OUT];
    __shared__ float red[256];
    __shared__ float inv_ssw_s, inv_scw_s;

    const int tid = threadIdx.x;
    const int bid = blockIdx.x;                    // B_*(H_/TH)*(W_/TW) = 4096
    const int b   = bid >> 9;                      // / (32*16)
    const int th_ = (bid >> 4) & 31;
    const int tw_ = bid & 15;
    const int h0  = th_ * TH;
    const int w0  = tw_ * TW;

    // ---- phase 0: weights (softplus) + sums ---------------------------------
    if (tid < C_IN) {
        wp_s[tid]   = softplus_f(w_prop[tid]);
        bias_s[tid] = bias[tid];
    }
    for (int i = tid; i < C_IN * 25; i += 256)
        sw_s[i] = softplus_f(spatial_weight[i]);
    for (int i = tid; i < C_OUT * C_IN; i += 256) {
        int o = i >> 5, c = i & 31;
        cw_s[o * 33 + c] = softplus_f(channel_weight[i]);
    }
    __syncthreads();

    float ps = 0.0f;
    for (int i = tid; i < C_IN * 25; i += 256) ps += sw_s[i];
    red[tid] = ps;
    __syncthreads();
    for (int s = 128; s > 0; s >>= 1) {
        if (tid < s) red[tid] += red[tid + s];
        __syncthreads();
    }
    if (tid == 0) inv_ssw_s = 1.0f / red[0];
    __syncthreads();

    float pc = 0.0f;
    for (int i = tid; i < C_OUT * C_IN; i += 256) pc += cw_s[(i >> 5) * 33 + (i & 31)];
    red[tid] = pc;
    __syncthreads();
    for (int s = 128; s > 0; s >>= 1) {
        if (tid < s) red[tid] += red[tid + s];
        __syncthreads();
    }
    if (tid == 0) inv_scw_s = 1.0f / red[0];
    __syncthreads();

    // ---- phase 1: elementwise prop stage into halo LDS ----------------------
    // Reference's pad construction => d_left[w] = d[w] masked at w==W-1,
    // d_right[w] = d[w] masked at w==0 (same w index, no shift!).
    #pragma unroll 2
    for (int it = 0; it < (C_IN * HALO_N) / 256; ++it) {   // 30 iterations
        int e  = tid + it * 256;
        int c  = e / HALO_N;
        int r  = e - c * HALO_N;
        int hh = r / HALO_W;
        int ww = r - hh * HALO_W;
        int gh = h0 - 2 + hh;
        int gw = w0 - 2 + ww;
        float a = 0.0f, cp = 0.0f;
        if ((unsigned)gh < (unsigned)H_ && (unsigned)gw < (unsigned)W_) {
            size_t idx = (((size_t)b * C_IN + c) * H_ + gh) * W_ + gw;
            float dv  = d[idx];
            float cdv = cd[idx];
            float gv  = gx[idx];
            float cgv = cgx[idx];
            float d_l  = (gw < W_ - 1) ? dv  : 0.0f;
            float cd_l = (gw < W_ - 1) ? cdv : 0.0f;
            float d_r  = (gw > 0)      ? dv  : 0.0f;
            float cd_r = (gw > 0)      ? cdv : 0.0f;
            float cgx_ds = cd_l * cd_r;
            float height = (cd_l * d_l + cd_r * d_r) / (cd_l + cd_r + EPSF);
            float gx_ds  = (d_r - d_l) * 0.5f / (height + EPSF);
            float wp  = wp_s[c];
            float gxp = (wp * cgv * gv + cgx_ds * gx_ds) / (wp * cgv + cgx_ds + EPSF);
            float cgp = (wp * cgv + cgx_ds) / (wp + 1.0f);
            a  = cgp * gxp;
            cp = cgp;
        }
        haloA[c * HALO_N + r] = a;
        haloC[c * HALO_N + r] = cp;
    }
    __syncthreads();

    // ---- phase 2: depthwise 5x5 conv -> X panel [c][pixel] ------------------
    #pragma unroll 2
    for (int it = 0; it < 16; ++it) {              // 32*8*16 / 256
        int e  = tid + it * 256;
        int c  = e >> 7;
        int r  = e & 127;                          // ph*16+pw
        int ph = r >> 4;
        int pw = r & 15;
        const float* swc = &sw_s[c * 25];
        const float* hA  = &haloA[c * HALO_N + ph * HALO_W + pw];
        const float* hC  = &haloC[c * HALO_N + ph * HALO_W + pw];
        float nom = 0.0f, den = 0.0f;
        #pragma unroll
        for (int i = 0; i < 5; ++i) {
            #pragma unroll
            for (int j = 0; j < 5; ++j) {
                float wv = swc[i * 5 + j];
                nom = fmaf(wv, hA[i * HALO_W + j], nom);
                den = fmaf(wv, hC[i * HALO_W + j], den);
            }
        }
        float gxs = nom / (den + EPSF);
        float cgs = den * inv_ssw_s;
        X[c * XSTRIDE + r]       = cgs * gxs;      // nom operand of 1x1 conv
        X[c * XSTRIDE + 128 + r] = cgs;            // denom operand
    }
    __syncthreads();

    // ---- phase 3: 1x1 channel conv as WMMA GEMM (32 x 256 x 32) -------------
    // v_wmma_f32_16x16x4_f32: A 16x4 (M=o, K=c), B 4x16 (K=c, N=pixel-col).
    // ISA A layout: lanes 0-15 -> K={0,1}, lanes 16-31 -> K={2,3}; M = lane%16.
    const int lane = tid & 31;
    const int wv   = tid >> 5;          // 0..7
    const int half = lane >> 4;
    const int l16  = lane & 15;
    const int m    = wv >> 2;           // output-channel tile: o in [16m,16m+16)
    const int ntB  = (wv & 3) * 2;      // this wave's two row-tiles

    v2f afrag[8];
    #pragma unroll
    for (int k = 0; k < 8; ++k) {       // K chunk: channels 4k..4k+3
        int base = (m * 16 + l16) * 33 + k * 4 + 2 * half;
        afrag[k].x = cw_s[base];
        afrag[k].y = cw_s[base + 1];
    }

    #pragma unroll
    for (int pi = 0; pi < 2; ++pi) {
        const int nt = ntB + pi;        // tile row (8 rows of 16 pixels)
        v8f accN = {};
        v8f accD = {};
        #pragma unroll
        for (int k = 0; k < 8; ++k) {
            int xr = (k * 4 + 2 * half) * XSTRIDE + l16;
            v2f bN, bD;
            bN.x = X[xr + nt * 16];
            bN.y = X[xr + XSTRIDE + nt * 16];
            bD.x = X[xr + 128 + nt * 16];
            bD.y = X[xr + XSTRIDE + 128 + nt * 16];
            accN = __builtin_amdgcn_wmma_f32_16x16x4_f32(
                false, afrag[k], false, bN, (short)0, accN, false, false);
            accD = __builtin_amdgcn_wmma_f32_16x16x4_f32(
                false, afrag[k], false, bD, (short)0, accD, false, false);
        }
        // epilogue: gx_out = nom/(den+eps)+bias ; cgx_out = den/sum(cw)
        const int gh  = h0 + nt;
        const int gw  = w0 + l16;
        #pragma unroll
        for (int r8 = 0; r8 < 8; ++r8) {
            int o = m * 16 + half * 8 + r8;       // D row M = r8 + 8*half
            float nomv = accN[r8];
            float denv = accD[r8];
            float gout = nomv / (denv + EPSF) + bias_s[o];
            float cout = denv * inv_scw_s;
            size_t oidx = (((size_t)b * C_OUT + o) * H_ + gh) * W_ + gw;
            out[oidx] = gout;
            out[oidx + (size_t)B_ * C_OUT * H_ * W_] = cout;
        }
    }
}

extern "C" void kernel_launch(void* const* d_in, const int* in_sizes, int n_in,
                              void* d_out, int out_size, void* d_ws, size_t ws_size,
                              hipStream_t stream) {
    (void)in_sizes; (void)n_in; (void)d_ws; (void)ws_size; (void)out_size;
    const float* d    = (const float*)d_in[0];
    const float* cd   = (const float*)d_in[1];
    const float* gx   = (const float*)d_in[2];
    const float* cgx  = (const float*)d_in[3];
    const float* wp   = (const float*)d_in[4];
    const float* sw   = (const float*)d_in[5];
    const float* cw   = (const float*)d_in[6];
    const float* bias = (const float*)d_in[7];
    float* out = (float*)d_out;

    const int grid = B_ * (H_ / TH) * (W_ / TW);   // 4096 workgroups
    const size_t shmem = (size_t)(2 * C_IN * HALO_N + C_IN * XSTRIDE) * sizeof(float); // ~92 KB

    structnconv_gx_fused<<<dim3(grid), dim3(256), shmem, stream>>>(
        d, cd, gx, cgx, wp, sw, cw, bias, out);
}